// MambaCore_84945863180802
// MI455X (gfx1250) — compile-verified
//
#include <hip/hip_runtime.h>
#include <hip/hip_bf16.h>
#include <math.h>

#define BATCH 32
#define CHN   96
#define SEQ   4096
#define DIN   256
#define DD2   512
#define NST   16

typedef float v2f __attribute__((ext_vector_type(2)));
typedef float v8f __attribute__((ext_vector_type(8)));

static __device__ __forceinline__ float wave_sum32(float v) {
#pragma unroll
  for (int o = 16; o > 0; o >>= 1) v += __shfl_xor(v, o, 32);
  return v;
}

static __device__ __forceinline__ float silu_f(float x) {
  return x / (1.0f + __expf(-x));
}

// Async copy: 16 bytes per lane, global -> LDS, tracked on ASYNCcnt.
static __device__ __forceinline__ void async_ld_b128(unsigned lds_off, const void* g) {
  asm volatile("global_load_async_to_lds_b128 %0, %1, off"
               :: "v"(lds_off), "v"(g) : "memory");
}
static __device__ __forceinline__ void wait_async0() {
  asm volatile("s_wait_asynccnt 0" ::: "memory");
}

// ---------------------------------------------------------------------------
// Generic per-batch GEMM:  O[b] (M x SEQ) = W (M x K) @ X[b] (K x SEQ) + bias
// One wave computes one 16x16 output tile with V_WMMA_F32_16X16X4_F32.
// act: 0 = none, 1 = softplus
// ---------------------------------------------------------------------------
__global__ __launch_bounds__(256) void gemm_f32_wmma(
    const float* __restrict__ W, const float* __restrict__ X,
    float* __restrict__ O, const float* __restrict__ bias,
    int M, int K, int xbstride, int obstride, int act)
{
  const int tile = blockIdx.x * 8 + (threadIdx.x >> 5);
  const int lane = threadIdx.x & 31;
  const int half = lane >> 4;       // 0: K rows {0,1}, 1: K rows {2,3}
  const int lm   = lane & 15;

  const int ntm = M >> 4;
  const int ntn = SEQ >> 4;
  const int tilesPerB = ntm * ntn;
  const int bi  = tile / tilesPerB;
  const int rem = tile - bi * tilesPerB;
  const int mt  = rem / ntn;
  const int nt  = rem - mt * ntn;

  const float* __restrict__ wrow = W + (size_t)(mt * 16 + lm) * K + 2 * half;
  const float* __restrict__ xcol = X + (size_t)bi * xbstride
                                     + (size_t)(2 * half) * SEQ + nt * 16 + lm;

  v8f c = {0.f, 0.f, 0.f, 0.f, 0.f, 0.f, 0.f, 0.f};
  for (int k0 = 0; k0 < K; k0 += 4) {
    v2f a, b;
    a.x = wrow[k0];
    a.y = wrow[k0 + 1];
    b.x = xcol[(size_t)k0 * SEQ];
    b.y = xcol[(size_t)(k0 + 1) * SEQ];
    c = __builtin_amdgcn_wmma_f32_16x16x4_f32(
        false, a, false, b, (short)0, c, false, false);
  }

  float* __restrict__ Ob = O + (size_t)bi * obstride;
  const int col = nt * 16 + lm;
#pragma unroll
  for (int v = 0; v < 8; ++v) {
    const int row = mt * 16 + v + 8 * half;
    float val = c[v] + (bias ? bias[row] : 0.f);
    if (act == 1) val = (val > 20.f) ? val : log1pf(__expf(val));
    Ob[(size_t)row * SEQ + col] = val;
  }
}

// ---------------------------------------------------------------------------
// Depthwise causal conv1d (k=4) + bias + SiLU.  x_res = xz rows [0,256).
// ---------------------------------------------------------------------------
__global__ __launch_bounds__(256) void conv_silu_kernel(
    const float* __restrict__ xz, const float* __restrict__ cw,
    const float* __restrict__ cb, float* __restrict__ u)
{
  const int gid = blockIdx.x * 256 + threadIdx.x;      // BATCH*DIN*(SEQ/4)
  const int perB = DIN * (SEQ / 4);
  const int bi   = gid / perB;
  const int rem  = gid - bi * perB;
  const int dch  = rem / (SEQ / 4);
  const int l0   = (rem - dch * (SEQ / 4)) * 4;

  const float* __restrict__ xr = xz + (size_t)bi * DD2 * SEQ + (size_t)dch * SEQ;
  const float w0 = cw[dch * 4 + 0], w1 = cw[dch * 4 + 1];
  const float w2 = cw[dch * 4 + 2], w3 = cw[dch * 4 + 3];
  const float bb = cb[dch];

  float4 cur = *(const float4*)&xr[l0];
  float p0 = 0.f, p1 = 0.f, p2 = 0.f;
  if (l0 >= 4) {
    float4 pv = *(const float4*)&xr[l0 - 4];
    p0 = pv.y; p1 = pv.z; p2 = pv.w;
  }
  float win[7] = {p0, p1, p2, cur.x, cur.y, cur.z, cur.w};
  float o[4];
#pragma unroll
  for (int s = 0; s < 4; ++s) {
    float t = w0 * win[s] + w1 * win[s + 1] + w2 * win[s + 2] + w3 * win[s + 3] + bb;
    o[s] = silu_f(t);
  }
  float4 ov = {o[0], o[1], o[2], o[3]};
  *(float4*)&u[(size_t)bi * DIN * SEQ + (size_t)dch * SEQ + l0] = ov;
}

// ---------------------------------------------------------------------------
// Selective scan. One lane owns one (b,d) channel, 16 states in registers.
// B/C tiles (16x64) stream into double-buffered LDS via async-to-LDS DMA,
// overlapped with the recurrence of the current chunk.
// ---------------------------------------------------------------------------
__global__ __launch_bounds__(64) void scan_kernel(
    const float* __restrict__ dt, const float* __restrict__ u,
    const float* __restrict__ dtbc, const float* __restrict__ A_log,
    float* __restrict__ y)
{
  const int bi  = blockIdx.x >> 2;
  const int dg  = blockIdx.x & 3;
  const int d   = dg * 64 + threadIdx.x;
  const int tid = threadIdx.x;

  float A2[NST], h[NST];
#pragma unroll
  for (int n = 0; n < NST; ++n) {
    A2[n] = -__expf(A_log[d * NST + n]) * 1.44269504088896f;  // * log2(e)
    h[n]  = 0.f;
  }

  __shared__ float sB[2][NST][64];
  __shared__ float sC[2][NST][64];
  const unsigned sB0 = (unsigned)(uintptr_t)&sB[0][0][0];
  const unsigned sC0 = (unsigned)(uintptr_t)&sC[0][0][0];

  const float* __restrict__ dtrow = dt + (size_t)(bi * DIN + d) * SEQ;
  const float* __restrict__ urow  = u  + (size_t)(bi * DIN + d) * SEQ;
  float*       __restrict__ yrow  = y  + (size_t)(bi * DIN + d) * SEQ;
  const float* __restrict__ Bb = dtbc + (size_t)bi * 48 * SEQ + (size_t)16 * SEQ;
  const float* __restrict__ Cb = dtbc + (size_t)bi * 48 * SEQ + (size_t)32 * SEQ;

  // Each thread DMAs 4 float4 pieces of each 16x64 tile per buffer.
  auto issue_tiles = [&](int buf, int l0) {
#pragma unroll
    for (int it = 0; it < 4; ++it) {
      const int idx = it * 64 + tid;
      const int row = idx >> 4;
      const int col = (idx & 15) * 4;
      const unsigned off = (unsigned)((buf * NST * 64 + row * 64 + col) * 4);
      async_ld_b128(sB0 + off, (const void*)(Bb + (size_t)row * SEQ + l0 + col));
      async_ld_b128(sC0 + off, (const void*)(Cb + (size_t)row * SEQ + l0 + col));
    }
  };

  issue_tiles(0, 0);

  for (int cchunk = 0; cchunk < SEQ / 64; ++cchunk) {
    const int l0  = cchunk * 64;
    const int cur = cchunk & 1;

    wait_async0();        // this wave's DMA (incl. current buffer) complete
    __syncthreads();      // every wave's DMA visible; prev buffer free

    if (cchunk + 1 < SEQ / 64) {
      issue_tiles(cur ^ 1, l0 + 64);      // overlap next chunk's DMA
      __builtin_prefetch(dtrow + l0 + 64, 0, 0);
      __builtin_prefetch(urow + l0 + 64, 0, 0);
    }

    for (int li = 0; li < 64; li += 4) {
      float4 d4 = *(const float4*)&dtrow[l0 + li];
      float4 u4 = *(const float4*)&urow[l0 + li];
      float dd[4] = {d4.x, d4.y, d4.z, d4.w};
      float uu[4] = {u4.x, u4.y, u4.z, u4.w};
      float oo[4];
#pragma unroll
      for (int s = 0; s < 4; ++s) {
        const float dtv = dd[s];
        const float du  = dtv * uu[s];
        float yv = 0.f;
#pragma unroll
        for (int n = 0; n < NST; ++n) {
          const float dA = __builtin_amdgcn_exp2f(dtv * A2[n]);
          h[n] = dA * h[n] + du * sB[cur][n][li + s];
          yv += h[n] * sC[cur][n][li + s];
        }
        oo[s] = yv;
      }
      float4 ov = {oo[0], oo[1], oo[2], oo[3]};
      *(float4*)&yrow[l0 + li] = ov;
    }
  }
}

// ---------------------------------------------------------------------------
// v = (y + u*D) * silu(z) + x_res, transposed (d,L) -> token-major (L,d).
// ---------------------------------------------------------------------------
__global__ __launch_bounds__(256) void gate_transpose_kernel(
    const float* __restrict__ y, const float* __restrict__ u,
    const float* __restrict__ xz, const float* __restrict__ Dp,
    float* __restrict__ vt)
{
  const int bi = blockIdx.x >> 8;
  const int r  = blockIdx.x & 255;
  const int d0 = (r >> 6) * 64;
  const int l0 = (r & 63) * 64;
  const int tid = threadIdx.x;

  __shared__ float t[64][65];

#pragma unroll
  for (int it = 0; it < 16; ++it) {
    const int i  = it * 256 + tid;
    const int rr = i >> 6;        // d-local
    const int cc = i & 63;        // l-local
    const int dch = d0 + rr;
    const size_t base = (size_t)bi * DIN * SEQ + (size_t)dch * SEQ + l0 + cc;
    const float yv = y[base];
    const float uv = u[base];
    const float xr = xz[(size_t)bi * DD2 * SEQ + (size_t)dch * SEQ + l0 + cc];
    const float zv = xz[(size_t)bi * DD2 * SEQ + (size_t)(DIN + dch) * SEQ + l0 + cc];
    t[rr][cc] = (yv + uv * Dp[dch]) * silu_f(zv) + xr;
  }
  __syncthreads();
#pragma unroll
  for (int it = 0; it < 16; ++it) {
    const int i  = it * 256 + tid;
    const int rr = i >> 6;        // l-local
    const int cc = i & 63;        // d-local
    vt[((size_t)bi * SEQ + l0 + rr) * DIN + d0 + cc] = t[cc][rr];
  }
}

// ---------------------------------------------------------------------------
// LN1: one wave32 per token (256 features, 8 per lane).
// ---------------------------------------------------------------------------
__global__ __launch_bounds__(256) void ln1_kernel(
    const float* __restrict__ vt, const float* __restrict__ g,
    const float* __restrict__ b, float* __restrict__ y1)
{
  const int wave = threadIdx.x >> 5;
  const int lane = threadIdx.x & 31;
  const size_t tok = (size_t)blockIdx.x * 8 + wave;

  const float4* __restrict__ row = (const float4*)(vt + tok * DIN);
  float4 a0 = row[lane];
  float4 a1 = row[32 + lane];
  float s  = a0.x + a0.y + a0.z + a0.w + a1.x + a1.y + a1.z + a1.w;
  float sq = a0.x*a0.x + a0.y*a0.y + a0.z*a0.z + a0.w*a0.w +
             a1.x*a1.x + a1.y*a1.y + a1.z*a1.z + a1.w*a1.w;
  s  = wave_sum32(s);
  sq = wave_sum32(sq);
  const float m    = s * (1.0f / 256.0f);
  const float var  = sq * (1.0f / 256.0f) - m * m;
  const float rstd = rsqrtf(var + 1e-5f);

  const float4 g0 = ((const float4*)g)[lane],  g1 = ((const float4*)g)[32 + lane];
  const float4 b0 = ((const float4*)b)[lane],  b1 = ((const float4*)b)[32 + lane];
  float4 o0, o1;
  o0.x = (a0.x - m) * rstd * g0.x + b0.x;  o0.y = (a0.y - m) * rstd * g0.y + b0.y;
  o0.z = (a0.z - m) * rstd * g0.z + b0.z;  o0.w = (a0.w - m) * rstd * g0.w + b0.w;
  o1.x = (a1.x - m) * rstd * g1.x + b1.x;  o1.y = (a1.y - m) * rstd * g1.y + b1.y;
  o1.z = (a1.z - m) * rstd * g1.z + b1.z;  o1.w = (a1.w - m) * rstd * g1.w + b1.w;
  float4* __restrict__ out = (float4*)(y1 + tok * DIN);
  out[lane] = o0;
  out[32 + lane] = o1;
}

// ---------------------------------------------------------------------------
// out_proj (WMMA) + residual + LN2, fused per 16-token block.
// ---------------------------------------------------------------------------
__global__ __launch_bounds__(256) void outproj_ln2_kernel(
    const float* __restrict__ y1, const float* __restrict__ Wout,
    const float* __restrict__ outb, const float* __restrict__ g2,
    const float* __restrict__ b2, float* __restrict__ out)
{
  const int wave = threadIdx.x >> 5;
  const int lane = threadIdx.x & 31;
  const int half = lane >> 4;
  const int lm   = lane & 15;
  const size_t tok0 = (size_t)blockIdx.x * 16;
  const int dout0 = wave * 32;

  const float* __restrict__ Arow = y1 + (tok0 + lm) * DIN + 2 * half;
  const float* __restrict__ B0   = Wout + (size_t)(dout0 + lm) * DIN + 2 * half;
  const float* __restrict__ B1   = Wout + (size_t)(dout0 + 16 + lm) * DIN + 2 * half;

  v8f c0 = {0.f, 0.f, 0.f, 0.f, 0.f, 0.f, 0.f, 0.f};
  v8f c1 = {0.f, 0.f, 0.f, 0.f, 0.f, 0.f, 0.f, 0.f};
  for (int k0 = 0; k0 < DIN; k0 += 4) {
    v2f a, bb0, bb1;
    a.x   = Arow[k0]; a.y   = Arow[k0 + 1];
    bb0.x = B0[k0];   bb0.y = B0[k0 + 1];
    bb1.x = B1[k0];   bb1.y = B1[k0 + 1];
    c0 = __builtin_amdgcn_wmma_f32_16x16x4_f32(false, a, false, bb0, (short)0, c0, false, false);
    c1 = __builtin_amdgcn_wmma_f32_16x16x4_f32(false, a, false, bb1, (short)0, c1, false, false);
  }

  __shared__ float st[16][264];
#pragma unroll
  for (int v = 0; v < 8; ++v) {
    const int tokl = v + 8 * half;
    const int da = dout0 + lm;
    const int db = dout0 + 16 + lm;
    st[tokl][da] = c0[v] + outb[da] + y1[(tok0 + tokl) * DIN + da];
    st[tokl][db] = c1[v] + outb[db] + y1[(tok0 + tokl) * DIN + db];
  }
  __syncthreads();

#pragma unroll
  for (int tt = 0; tt < 2; ++tt) {
    const int tokl = wave * 2 + tt;
    const float* rowp = &st[tokl][0];
    float4 a0 = *(const float4*)&rowp[lane * 4];
    float4 a1 = *(const float4*)&rowp[128 + lane * 4];
    float s  = a0.x + a0.y + a0.z + a0.w + a1.x + a1.y + a1.z + a1.w;
    float sq = a0.x*a0.x + a0.y*a0.y + a0.z*a0.z + a0.w*a0.w +
               a1.x*a1.x + a1.y*a1.y + a1.z*a1.z + a1.w*a1.w;
    s  = wave_sum32(s);
    sq = wave_sum32(sq);
    const float m    = s * (1.0f / 256.0f);
    const float var  = sq * (1.0f / 256.0f) - m * m;
    const float rstd = rsqrtf(var + 1e-5f);

    const float4 gg0 = ((const float4*)g2)[lane], gg1 = ((const float4*)g2)[32 + lane];
    const float4 bb0 = ((const float4*)b2)[lane], bb1 = ((const float4*)b2)[32 + lane];
    float4 o0, o1;
    o0.x = (a0.x - m) * rstd * gg0.x + bb0.x;  o0.y = (a0.y - m) * rstd * gg0.y + bb0.y;
    o0.z = (a0.z - m) * rstd * gg0.z + bb0.z;  o0.w = (a0.w - m) * rstd * gg0.w + bb0.w;
    o1.x = (a1.x - m) * rstd * gg1.x + bb1.x;  o1.y = (a1.y - m) * rstd * gg1.y + bb1.y;
    o1.z = (a1.z - m) * rstd * gg1.z + bb1.z;  o1.w = (a1.w - m) * rstd * gg1.w + bb1.w;
    float4* op = (float4*)(out + (tok0 + tokl) * DIN);
    op[lane] = o0;
    op[32 + lane] = o1;
  }
}

// ---------------------------------------------------------------------------
extern "C" void kernel_launch(void* const* d_in, const int* in_sizes, int n_in,
                              void* d_out, int out_size, void* d_ws, size_t ws_size,
                              hipStream_t stream) {
  const float* x        = (const float*)d_in[0];
  const float* in_w     = (const float*)d_in[1];
  const float* in_b     = (const float*)d_in[2];
  const float* conv_w   = (const float*)d_in[3];
  const float* conv_b   = (const float*)d_in[4];
  const float* xproj_w  = (const float*)d_in[5];
  const float* dt_w     = (const float*)d_in[6];
  const float* dt_b     = (const float*)d_in[7];
  const float* out_w    = (const float*)d_in[8];
  const float* out_b    = (const float*)d_in[9];
  const float* A_log    = (const float*)d_in[10];
  const float* Dp       = (const float*)d_in[11];
  const float* ln1g     = (const float*)d_in[12];
  const float* ln1b     = (const float*)d_in[13];
  const float* ln2g     = (const float*)d_in[14];
  const float* ln2b     = (const float*)d_in[15];
  float* out = (float*)d_out;

  float* ws   = (float*)d_ws;
  float* xz   = ws;                                       // 32*512*4096
  float* u    = xz   + (size_t)BATCH * DD2 * SEQ;         // 32*256*4096
  float* dtbc = u    + (size_t)BATCH * DIN * SEQ;         // 32*48*4096
  float* dt   = dtbc + (size_t)BATCH * 48 * SEQ;          // 32*256*4096
  float* y    = dt   + (size_t)BATCH * DIN * SEQ;         // 32*256*4096
  float* vt   = dt;   // dt is dead after the scan -> reuse for token-major v
  float* y1   = y;    // y is dead after gate_transpose -> reuse for LN1 out

  // 1) in_proj: xz (b,512,L) = in_w(512x96) @ x_b(96xL) + in_b
  gemm_f32_wmma<<<BATCH * 32 * 256 / 8, 256, 0, stream>>>(
      in_w, x, xz, in_b, DD2, CHN, CHN * SEQ, DD2 * SEQ, 0);
  // 2) causal depthwise conv + SiLU -> u (b,256,L)
  conv_silu_kernel<<<BATCH * DIN * (SEQ / 4) / 256, 256, 0, stream>>>(
      xz, conv_w, conv_b, u);
  // 3) x_proj: dtbc (b,48,L) = xproj_w(48x256) @ u_b
  gemm_f32_wmma<<<BATCH * 3 * 256 / 8, 256, 0, stream>>>(
      xproj_w, u, dtbc, nullptr, 48, DIN, DIN * SEQ, 48 * SEQ, 0);
  // 4) dt: (b,256,L) = softplus(dt_w(256x16) @ dt_raw + dt_b)
  gemm_f32_wmma<<<BATCH * 16 * 256 / 8, 256, 0, stream>>>(
      dt_w, dtbc, dt, dt_b, DIN, 16, 48 * SEQ, DIN * SEQ, 1);
  // 5) selective scan -> y (b,256,L), async-DMA double-buffered B/C tiles
  scan_kernel<<<BATCH * 4, 64, 0, stream>>>(dt, u, dtbc, A_log, y);
  // 6) gate + residual + transpose -> vt (b*L, 256)
  gate_transpose_kernel<<<BATCH * 256, 256, 0, stream>>>(y, u, xz, Dp, vt);
  // 7) LN1 -> y1 (b*L, 256)
  ln1_kernel<<<BATCH * SEQ / 8, 256, 0, stream>>>(vt, ln1g, ln1b, y1);
  // 8) out_proj + residual + LN2 -> out (b,L,256)
  outproj_ln2_kernel<<<BATCH * SEQ / 16, 256, 0, stream>>>(
      y1, out_w, out_b, ln2g, ln2b, out);
}